// EIGENX_81398220194162
// MI455X (gfx1250) — compile-verified
//
#include <hip/hip_runtime.h>
#include <hip/hip_bf16.h>
#include <math.h>

#define NN   100000
#define EE   1600000
#define INC  128
#define HIDC 256
#define KEIG 64
#define OUTC 16

typedef __attribute__((ext_vector_type(2))) float v2f;
typedef __attribute__((ext_vector_type(8))) float v8f;

// ---------------------------------------------------------------------------
// 1) Transpose MLP_W [64, N] -> W_T [N, 64] (coalesced read, 256B-row writes)
// ---------------------------------------------------------------------------
__global__ void EIG_transpose(const float* __restrict__ W, float* __restrict__ WT) {
    long long idx = (long long)blockIdx.x * blockDim.x + threadIdx.x;
    if (idx >= (long long)NN * KEIG) return;
    int n = (int)(idx % NN);
    int k = (int)(idx / NN);
    WT[(size_t)n * KEIG + k] = W[(size_t)k * NN + n];
}

// ---------------------------------------------------------------------------
// 2) Edge scatter: dst[col[e]*64+k] += src[row[e]*64+k]; optional degree.
//    64 threads per edge -> contiguous 256B gather + contiguous atomic scatter.
//    src/dst tables are 25.6MB each -> resident in MI455X 192MB L2.
// ---------------------------------------------------------------------------
__global__ void EIG_spmm(const int* __restrict__ row, const int* __restrict__ col,
                         const float* __restrict__ src, float* __restrict__ dst,
                         float* __restrict__ deg) {
    long long idx = (long long)blockIdx.x * blockDim.x + threadIdx.x;
    if (idx >= (long long)EE * KEIG) return;
    int e = (int)(idx >> 6);
    int k = (int)(idx & 63);
    int r = row[e];
    int c = col[e];
    atomicAdd(&dst[(size_t)c * KEIG + k], src[(size_t)r * KEIG + k]);
    if (deg && k == 0) atomicAdd(&deg[c], 1.0f);
}

// ---------------------------------------------------------------------------
// 3) Bias + softmax over 64 channels; one wave32 per node, 2 elems/lane.
// ---------------------------------------------------------------------------
__global__ void EIG_softmax(float* __restrict__ xA, const float* __restrict__ b) {
    int gid  = blockIdx.x * blockDim.x + threadIdx.x;
    int node = gid >> 5;
    int lane = gid & 31;
    if (node >= NN) return;
    size_t base = (size_t)node * KEIG;
    float v0 = xA[base + lane]      + b[lane];
    float v1 = xA[base + 32 + lane] + b[32 + lane];
    float m = fmaxf(v0, v1);
    for (int off = 16; off >= 1; off >>= 1) m = fmaxf(m, __shfl_xor(m, off, 32));
    float e0 = expf(v0 - m), e1 = expf(v1 - m);
    float s = e0 + e1;
    for (int off = 16; off >= 1; off >>= 1) s += __shfl_xor(s, off, 32);
    float inv = 1.0f / s;
    xA[base + lane]      = e0 * inv;
    xA[base + 32 + lane] = e1 * inv;
}

// ---------------------------------------------------------------------------
// 4) cut = sum(S*AS), deg_term = sum(deg[n]*S^2) -- grid-stride + LDS reduce
// ---------------------------------------------------------------------------
__global__ void EIG_cutdeg(const float* __restrict__ S, const float* __restrict__ AS,
                           const float* __restrict__ deg, float* __restrict__ scal) {
    __shared__ float rc[256], rd[256];
    const int total = NN * KEIG;
    float c = 0.f, d = 0.f;
    for (int i = blockIdx.x * blockDim.x + threadIdx.x; i < total;
         i += gridDim.x * blockDim.x) {
        float s = S[i];
        c += s * AS[i];
        d += deg[i >> 6] * s * s;
    }
    rc[threadIdx.x] = c; rd[threadIdx.x] = d; __syncthreads();
    for (int s2 = 128; s2 > 0; s2 >>= 1) {
        if ((int)threadIdx.x < s2) {
            rc[threadIdx.x] += rc[threadIdx.x + s2];
            rd[threadIdx.x] += rd[threadIdx.x + s2];
        }
        __syncthreads();
    }
    if (threadIdx.x == 0) { atomicAdd(&scal[0], rc[0]); atomicAdd(&scal[1], rd[0]); }
}

// ---------------------------------------------------------------------------
// 5) SS = S^T S  [64x64], K=N.  f32 WMMA 16x16x4, 16 tiles x 50 K-splits,
//    atomic accumulate partials into SS.
// ---------------------------------------------------------------------------
__global__ void EIG_ss_wmma(const float* __restrict__ S, float* __restrict__ SS) {
    int wid  = (blockIdx.x * blockDim.x + threadIdx.x) >> 5;   // 0..799
    int lane = threadIdx.x & 31;
    if (wid >= 800) return;
    int tile = wid & 15;          // 16 output tiles of 16x16
    int ks   = wid >> 4;          // 50 K-splits of 2000 nodes
    int ti = (tile >> 2) * 16;    // row block (cols of S)
    int tj = (tile & 3) * 16;     // col block (cols of S)
    int mi = lane & 15;
    int ko = (lane & 16) ? 2 : 0; // A/B half-wave K offset per ISA layout
    v8f c = {};
    int node0 = ks * 2000;
    for (int nb = 0; nb < 2000; nb += 4) {
        int node = node0 + nb;
        v2f a, b;
        a.x = S[(size_t)(node + ko)     * KEIG + ti + mi];
        a.y = S[(size_t)(node + ko + 1) * KEIG + ti + mi];
        b.x = S[(size_t)(node + ko)     * KEIG + tj + mi];
        b.y = S[(size_t)(node + ko + 1) * KEIG + tj + mi];
        c = __builtin_amdgcn_wmma_f32_16x16x4_f32(false, a, false, b,
                                                  (short)0, c, false, false);
    }
    int mrow = (lane & 16) ? 8 : 0;
    for (int v = 0; v < 8; ++v)
        atomicAdd(&SS[(size_t)(ti + v + mrow) * KEIG + tj + mi], c[v]);
}

// ---------------------------------------------------------------------------
// 6) Finalize loss: ||SS/||SS||_F - I/8||_F  - cut/deg_term -> d_out[N*16]
// ---------------------------------------------------------------------------
__global__ void EIG_finalize(const float* __restrict__ SS, const float* __restrict__ scal,
                             float* __restrict__ out_loss) {
    __shared__ float red[256];
    __shared__ float nf;
    int t = threadIdx.x;
    float a = 0.f;
    for (int i = t; i < KEIG * KEIG; i += 256) { float v = SS[i]; a += v * v; }
    red[t] = a; __syncthreads();
    for (int s = 128; s > 0; s >>= 1) { if (t < s) red[t] += red[t + s]; __syncthreads(); }
    if (t == 0) nf = sqrtf(red[0]);
    __syncthreads();
    float b = 0.f;
    for (int i = t; i < KEIG * KEIG; i += 256) {
        int r = i >> 6, cc = i & 63;
        float v = SS[i] / nf - (r == cc ? 0.125f : 0.0f);  // 1/sqrt(64)=0.125
        b += v * v;
    }
    red[t] = b; __syncthreads();
    for (int s = 128; s > 0; s >>= 1) { if (t < s) red[t] += red[t + s]; __syncthreads(); }
    if (t == 0) {
        float ortho = sqrtf(red[0]);
        float pump  = -(scal[0] / scal[1]);
        out_loss[0] = pump + ortho;
    }
}

// ---------------------------------------------------------------------------
// 7) xX = x @ mlpx_W.T + b   [N,128]x[128,256], f32 WMMA, 1 wave = 16x16 tile
// ---------------------------------------------------------------------------
__global__ void EIG_gemm_xX(const float* __restrict__ x, const float* __restrict__ W,
                            const float* __restrict__ bias, float* __restrict__ xX) {
    int wid  = (blockIdx.x * blockDim.x + threadIdx.x) >> 5;
    int lane = threadIdx.x & 31;
    if (wid >= (NN / 16) * (HIDC / 16)) return;
    int ti = wid / (HIDC / 16);
    int tj = wid % (HIDC / 16);
    int row0 = ti * 16, col0 = tj * 16;
    int mi = lane & 15;
    int ko = (lane & 16) ? 2 : 0;
    v8f c = {};
    for (int k = 0; k < INC; k += 4) {
        int kk = k + ko;
        v2f a, b;
        a.x = x[(size_t)(row0 + mi) * INC + kk];
        a.y = x[(size_t)(row0 + mi) * INC + kk + 1];
        // B(k,n) = mlpx_W.T[k][n] = mlpx_W[n*128 + k]
        b.x = W[(size_t)(col0 + mi) * INC + kk];
        b.y = W[(size_t)(col0 + mi) * INC + kk + 1];
        c = __builtin_amdgcn_wmma_f32_16x16x4_f32(false, a, false, b,
                                                  (short)0, c, false, false);
    }
    int mrow = (lane & 16) ? 8 : 0;
    float bv = bias[col0 + mi];
    for (int v = 0; v < 8; ++v)
        xX[(size_t)(row0 + v + mrow) * HIDC + col0 + mi] = c[v] + bv;
}

// ---------------------------------------------------------------------------
// 8) z = log_softmax(concat(xX,S) @ fin_W.T + fin_b)  [N,320]x[320,16]
//    f32 WMMA, 1 wave = 16 rows; fused log-softmax via 16-lane shfl groups.
// ---------------------------------------------------------------------------
__global__ void EIG_gemm_final(const float* __restrict__ xX, const float* __restrict__ S,
                               const float* __restrict__ W, const float* __restrict__ bias,
                               float* __restrict__ out) {
    int wid  = (blockIdx.x * blockDim.x + threadIdx.x) >> 5;
    int lane = threadIdx.x & 31;
    if (wid >= NN / 16) return;
    int row0 = wid * 16;
    int mi = lane & 15;
    int ko = (lane & 16) ? 2 : 0;
    int rowA = row0 + mi;
    v8f c = {};
    for (int k = 0; k < HIDC + KEIG; k += 4) {
        int kk = k + ko;
        v2f a, b;
        a.x = (kk     < HIDC) ? xX[(size_t)rowA * HIDC + kk]
                              : S[(size_t)rowA * KEIG + (kk - HIDC)];
        a.y = (kk + 1 < HIDC) ? xX[(size_t)rowA * HIDC + kk + 1]
                              : S[(size_t)rowA * KEIG + (kk + 1 - HIDC)];
        b.x = W[(size_t)mi * (HIDC + KEIG) + kk];
        b.y = W[(size_t)mi * (HIDC + KEIG) + kk + 1];
        c = __builtin_amdgcn_wmma_f32_16x16x4_f32(false, a, false, b,
                                                  (short)0, c, false, false);
    }
    int mrow = (lane & 16) ? 8 : 0;
    float bv = bias[mi];
    for (int v = 0; v < 8; ++v) {
        float val = c[v] + bv;
        // each 16-lane half-wave holds one full output row of 16 logits
        float m = val;
        for (int off = 8; off >= 1; off >>= 1) m = fmaxf(m, __shfl_xor(m, off, 32));
        float e = expf(val - m);
        float s = e;
        for (int off = 8; off >= 1; off >>= 1) s += __shfl_xor(s, off, 32);
        out[(size_t)(row0 + v + mrow) * OUTC + mi] = (val - m) - logf(s);
    }
}

// ---------------------------------------------------------------------------
extern "C" void kernel_launch(void* const* d_in, const int* in_sizes, int n_in,
                              void* d_out, int out_size, void* d_ws, size_t ws_size,
                              hipStream_t stream) {
    const float* x      = (const float*)d_in[0];
    const int*   eidx   = (const int*)d_in[1];
    const float* MLP_W  = (const float*)d_in[2];
    const float* MLP_b  = (const float*)d_in[3];
    const float* mlpx_W = (const float*)d_in[4];
    const float* mlpx_b = (const float*)d_in[5];
    const float* fin_W  = (const float*)d_in[6];
    const float* fin_b  = (const float*)d_in[7];
    float* z_out = (float*)d_out;

    const int* row = eidx;
    const int* col = eidx + EE;

    // workspace layout (256B aligned)
    char* base = (char*)d_ws;
    size_t off = 0;
    auto take = [&](size_t bytes) -> void* {
        void* p = base + off;
        off = (off + bytes + 255) & ~(size_t)255;
        return p;
    };
    float* W_T   = (float*)take((size_t)NN * KEIG * 4);   // 25.6 MB
    float* xA    = (float*)take((size_t)NN * KEIG * 4);   // 25.6 MB (becomes S)
    float* AS    = (float*)take((size_t)NN * KEIG * 4);   // 25.6 MB
    float* xX    = (float*)take((size_t)NN * HIDC * 4);   // 102.4 MB
    float* deg   = (float*)take((size_t)NN * 4);          // 0.4 MB
    float* SS    = (float*)take((size_t)KEIG * KEIG * 4); // 16 KB
    float* scal  = (float*)take(256);                     // cut, deg_term

    // zero accumulators (graph-capture-safe memset nodes)
    hipMemsetAsync(xA,   0, (size_t)NN * KEIG * 4, stream);
    hipMemsetAsync(AS,   0, (size_t)NN * KEIG * 4, stream);
    hipMemsetAsync(deg,  0, (size_t)NN * 4, stream);
    hipMemsetAsync(SS,   0, (size_t)KEIG * KEIG * 4, stream);
    hipMemsetAsync(scal, 0, 256, stream);

    // 1) transpose weight table for contiguous edge gathers
    {
        long long tot = (long long)NN * KEIG;
        int blocks = (int)((tot + 255) / 256);
        EIG_transpose<<<blocks, 256, 0, stream>>>(MLP_W, W_T);
    }
    // 2) xA = A @ MLP_W.T (edge scatter) + degree
    {
        long long tot = (long long)EE * KEIG;
        int blocks = (int)((tot + 255) / 256);
        EIG_spmm<<<blocks, 256, 0, stream>>>(row, col, W_T, xA, deg);
    }
    // 3) S = softmax(xA + b) in place
    EIG_softmax<<<(NN * 32 + 255) / 256, 256, 0, stream>>>(xA, MLP_b);
    // 4) AS = A @ S
    {
        long long tot = (long long)EE * KEIG;
        int blocks = (int)((tot + 255) / 256);
        EIG_spmm<<<blocks, 256, 0, stream>>>(row, col, xA, AS, nullptr);
    }
    // 5) cut / deg_term scalars
    EIG_cutdeg<<<4096, 256, 0, stream>>>(xA, AS, deg, scal);
    // 6) SS = S^T S via WMMA
    EIG_ss_wmma<<<100, 256, 0, stream>>>(xA, SS);
    // 7) loss scalar -> d_out[N*16]
    EIG_finalize<<<1, 256, 0, stream>>>(SS, scal, z_out + (size_t)NN * OUTC);
    // 8) xX GEMM via WMMA
    {
        int waves = (NN / 16) * (HIDC / 16);     // 100000 waves
        EIG_gemm_xX<<<(waves * 32 + 255) / 256, 256, 0, stream>>>(x, mlpx_W, mlpx_b, xX);
    }
    // 9) final GEMM + log_softmax via WMMA
    {
        int waves = NN / 16;                     // 6250 waves
        EIG_gemm_final<<<(waves * 32 + 255) / 256, 256, 0, stream>>>(xX, xA, fin_W, fin_b, z_out);
    }
}